// attnPool_9534827397794
// MI455X (gfx1250) — compile-verified
//
#include <hip/hip_runtime.h>
#include <hip/hip_bf16.h>
#include <math.h>

typedef float v2f __attribute__((ext_vector_type(2)));
typedef float v8f __attribute__((ext_vector_type(8)));

#define N_NODES 8192
#define C_DIM   512
#define K_SEL   2048

// ---------- wave32 reductions ----------
// Reduce across a 16-lane half (xor of bits 0..3 never crosses bit 4) — wave32 safe.
__device__ __forceinline__ float half16_max(float v) {
    v = fmaxf(v, __shfl_xor(v, 1, 32));
    v = fmaxf(v, __shfl_xor(v, 2, 32));
    v = fmaxf(v, __shfl_xor(v, 4, 32));
    v = fmaxf(v, __shfl_xor(v, 8, 32));
    return v;
}
__device__ __forceinline__ float half16_sum(float v) {
    v += __shfl_xor(v, 1, 32);
    v += __shfl_xor(v, 2, 32);
    v += __shfl_xor(v, 4, 32);
    v += __shfl_xor(v, 8, 32);
    return v;
}
__device__ __forceinline__ float wave32_sum(float v) {
    v += __shfl_xor(v, 16, 32);
    v += __shfl_xor(v, 8, 32);
    v += __shfl_xor(v, 4, 32);
    v += __shfl_xor(v, 2, 32);
    v += __shfl_xor(v, 1, 32);
    return v;
}

__device__ __forceinline__ v8f wmma_f32(v2f a, v2f b, v8f c) {
    return __builtin_amdgcn_wmma_f32_16x16x4_f32(
        false, a, false, b, (short)0, c, false, false);
}

// ---------------------------------------------------------------------------
// Kernel 1: K = X @ Wk_w^T + Wk_b      [8192,512] = [8192,512]x[512,512]^T
// One wave computes a 16x64 output strip (4 N-tiles): the A fragment is loaded
// once per d-step and reused for 4 WMMAs -> 1.25 loads/WMMA instead of 2.
// ISA f32 A-frag (16x4): lanes 0-15 -> K={0,1}, lanes 16-31 -> K={2,3}.
// B-frag (4x16) mirrors it; B[d][n] = Wk[n*512 + d] (contiguous v2f per lane).
// ---------------------------------------------------------------------------
__global__ void __launch_bounds__(32)
wk_gemm_kernel(const float* __restrict__ X, const float* __restrict__ Wk,
               const float* __restrict__ Wkb, float* __restrict__ Kout) {
    const int m0   = blockIdx.x * 16;
    const int n0   = blockIdx.y * 64;          // 4 tiles of 16 columns
    const int lane = threadIdx.x;
    const int half = lane >> 4;
    const int lr   = lane & 15;

    const float* arow = X  + (size_t)(m0 + lr) * C_DIM + half * 2;
    const float* b0   = Wk + (size_t)(n0 + lr)      * C_DIM + half * 2;
    const float* b1   = b0 + 16 * C_DIM;
    const float* b2   = b0 + 32 * C_DIM;
    const float* b3   = b0 + 48 * C_DIM;

    v8f acc0 = {}, acc1 = {}, acc2 = {}, acc3 = {};
#pragma unroll 4
    for (int d = 0; d < C_DIM; d += 4) {
        v2f a  = *reinterpret_cast<const v2f*>(arow + d);
        v2f f0 = *reinterpret_cast<const v2f*>(b0 + d);
        v2f f1 = *reinterpret_cast<const v2f*>(b1 + d);
        v2f f2 = *reinterpret_cast<const v2f*>(b2 + d);
        v2f f3 = *reinterpret_cast<const v2f*>(b3 + d);
        acc0 = wmma_f32(a, f0, acc0);
        acc1 = wmma_f32(a, f1, acc1);
        acc2 = wmma_f32(a, f2, acc2);
        acc3 = wmma_f32(a, f3, acc3);
    }
    const int mbase = m0 + half * 8;           // C/D layout: half1 -> rows M=8..15
    const float bi0 = Wkb[n0 + lr];
    const float bi1 = Wkb[n0 + 16 + lr];
    const float bi2 = Wkb[n0 + 32 + lr];
    const float bi3 = Wkb[n0 + 48 + lr];
#pragma unroll
    for (int r = 0; r < 8; ++r) {
        float* orow = Kout + (size_t)(mbase + r) * C_DIM + n0 + lr;
        orow[0]  = acc0[r] + bi0;
        orow[16] = acc1[r] + bi1;
        orow[32] = acc2[r] + bi2;
        orow[48] = acc3[r] + bi3;
    }
}

// ---------------------------------------------------------------------------
// Kernel 2: V[n] = dot(X[n], Wv_w) + Wv_b   (wave-per-row GEMV)
// ---------------------------------------------------------------------------
__global__ void __launch_bounds__(256)
wv_gemv_kernel(const float* __restrict__ X, const float* __restrict__ Wv,
               const float* __restrict__ Wvb, float* __restrict__ V) {
    const int wave = threadIdx.x >> 5;
    const int lane = threadIdx.x & 31;
    const int row  = blockIdx.x * (blockDim.x >> 5) + wave;
    if (row >= N_NODES) return;
    const float* xr = X + (size_t)row * C_DIM;
    float s = 0.f;
    for (int d = lane; d < C_DIM; d += 32) s += xr[d] * Wv[d];
    s = wave32_sum(s);
    if (lane == 0) V[row] = s + Wvb[0];
}

// ---------------------------------------------------------------------------
// Kernel 3: fused scores + online softmax + y = softmax(X K^T) @ V
// 16-row block per workgroup; 8 waves stride over 64-column groups (4 tiles).
// Per group: 512 f32 WMMAs sharing one A-frag load per d-step, then ONE
// online-softmax update over all 64 columns (4x fewer shuffles per WMMA).
// Wave partials merged via LDS log-sum-exp merge.
// ---------------------------------------------------------------------------
__global__ void __launch_bounds__(256)
attn_y_kernel(const float* __restrict__ X, const float* __restrict__ Kmat,
              const float* __restrict__ V, float* __restrict__ y) {
    const int i0   = blockIdx.x * 16;
    const int wave = threadIdx.x >> 5;     // 0..7
    const int lane = threadIdx.x & 31;
    const int half = lane >> 4;
    const int lr   = lane & 15;

    float m[8], l[8], ac[8];
#pragma unroll
    for (int r = 0; r < 8; ++r) { m[r] = -1e30f; l[r] = 0.f; ac[r] = 0.f; }

    const float* arow = X + (size_t)(i0 + lr) * C_DIM + half * 2;

    for (int jg = wave; jg < N_NODES / 64; jg += 8) {
        const int j0 = jg * 64;
        const float* b0 = Kmat + (size_t)(j0 + lr) * C_DIM + half * 2;
        const float* b1 = b0 + 16 * C_DIM;
        const float* b2 = b0 + 32 * C_DIM;
        const float* b3 = b0 + 48 * C_DIM;

        v8f s0 = {}, s1 = {}, s2 = {}, s3 = {};
#pragma unroll 4
        for (int d = 0; d < C_DIM; d += 4) {
            v2f a  = *reinterpret_cast<const v2f*>(arow + d);
            v2f f0 = *reinterpret_cast<const v2f*>(b0 + d);
            v2f f1 = *reinterpret_cast<const v2f*>(b1 + d);
            v2f f2 = *reinterpret_cast<const v2f*>(b2 + d);
            v2f f3 = *reinterpret_cast<const v2f*>(b3 + d);
            s0 = wmma_f32(a, f0, s0);
            s1 = wmma_f32(a, f1, s1);
            s2 = wmma_f32(a, f2, s2);
            s3 = wmma_f32(a, f3, s3);
        }

        const float vj0 = V[j0 + lr];
        const float vj1 = V[j0 + 16 + lr];
        const float vj2 = V[j0 + 32 + lr];
        const float vj3 = V[j0 + 48 + lr];
#pragma unroll
        for (int r = 0; r < 8; ++r) {
            const float a0 = s0[r], a1 = s1[r], a2 = s2[r], a3 = s3[r];
            const float tm = half16_max(fmaxf(fmaxf(a0, a1), fmaxf(a2, a3)));
            const float nm = fmaxf(m[r], tm);
            const float p0 = __expf(a0 - nm);
            const float p1 = __expf(a1 - nm);
            const float p2 = __expf(a2 - nm);
            const float p3 = __expf(a3 - nm);
            const float ps  = half16_sum((p0 + p1) + (p2 + p3));
            const float pvs = half16_sum((p0 * vj0 + p1 * vj1) + (p2 * vj2 + p3 * vj3));
            const float sc  = __expf(m[r] - nm);        // exp(-huge)=0 first time
            l[r]  = l[r]  * sc + ps;
            ac[r] = ac[r] * sc + pvs;
            m[r]  = nm;
        }
    }

    __shared__ float sm[8][16], sl[8][16], sa[8][16];
    if (lr == 0) {                       // lane 0 -> rows 0..7, lane 16 -> rows 8..15
#pragma unroll
        for (int r = 0; r < 8; ++r) {
            sm[wave][half * 8 + r] = m[r];
            sl[wave][half * 8 + r] = l[r];
            sa[wave][half * 8 + r] = ac[r];
        }
    }
    __syncthreads();

    if (threadIdx.x < 16) {
        const int R = threadIdx.x;
        float M = -1e30f;
#pragma unroll
        for (int w = 0; w < 8; ++w) M = fmaxf(M, sm[w][R]);
        float L = 0.f, A = 0.f;
#pragma unroll
        for (int w = 0; w < 8; ++w) {
            const float e = __expf(sm[w][R] - M);
            L += sl[w][R] * e;
            A += sa[w][R] * e;
        }
        y[i0 + R] = A / L;
    }
}

// ---------------------------------------------------------------------------
// Kernel 4: exact top-k rank. Strict total order (desc y, asc index) ->
// ranks are a permutation of [0,N); exactly K_SEL elements have rank < K_SEL.
// ---------------------------------------------------------------------------
__global__ void __launch_bounds__(256)
rank_kernel(const float* __restrict__ y, int* __restrict__ rank) {
    const int i = blockIdx.x * blockDim.x + threadIdx.x;
    if (i >= N_NODES) return;
    const float yi = y[i];
    int r = 0;
    for (int j = 0; j < N_NODES; ++j) {
        const float yj = y[j];
        r += (yj > yi) || (yj == yi && j < i);
    }
    rank[i] = r;
}

// ---------------------------------------------------------------------------
// Kernel 5: compact selected indices in ascending index order.
// pos(i) = #{ j < i : rank[j] < K_SEL }  -> sorted-unique semantics match ref.
// ---------------------------------------------------------------------------
__global__ void __launch_bounds__(256)
compact_kernel(const int* __restrict__ rank, int* __restrict__ idx,
               float* __restrict__ out_idx) {
    const int i = blockIdx.x * blockDim.x + threadIdx.x;
    if (i >= N_NODES) return;
    if (rank[i] < K_SEL) {
        int pos = 0;
        for (int j = 0; j < i; ++j) pos += (rank[j] < K_SEL);
        idx[pos]     = i;
        out_idx[pos] = (float)i;
    }
}

// ---------------------------------------------------------------------------
// Kernel 6: X_pooled = X[idx]   [2048, 512]
// ---------------------------------------------------------------------------
__global__ void __launch_bounds__(256)
gather_x_kernel(const float* __restrict__ X, const int* __restrict__ idx,
                float* __restrict__ Xp) {
    const int r   = blockIdx.x;
    const int src = idx[r];
    for (int c = threadIdx.x; c < C_DIM; c += blockDim.x)
        Xp[(size_t)r * C_DIM + c] = X[(size_t)src * C_DIM + c];
}

// ---------------------------------------------------------------------------
// Kernel 7: A_pooled = A[idx][:, idx]   [2048, 2048]
// ---------------------------------------------------------------------------
__global__ void __launch_bounds__(256)
gather_a_kernel(const float* __restrict__ A, const int* __restrict__ idx,
                float* __restrict__ Ap) {
    const int r = blockIdx.y;
    const int c = blockIdx.x * blockDim.x + threadIdx.x;
    if (c >= K_SEL) return;
    const int sr = idx[r];
    const int sc = idx[c];
    Ap[(size_t)r * K_SEL + c] = A[(size_t)sr * N_NODES + sc];
}

// ---------------------------------------------------------------------------
extern "C" void kernel_launch(void* const* d_in, const int* in_sizes, int n_in,
                              void* d_out, int out_size, void* d_ws, size_t ws_size,
                              hipStream_t stream) {
    const float* A    = (const float*)d_in[0];   // [8192, 8192]
    const float* X    = (const float*)d_in[1];   // [8192, 512]
    const float* Wk_w = (const float*)d_in[2];   // [512, 512]
    const float* Wk_b = (const float*)d_in[3];   // [512]
    const float* Wv_w = (const float*)d_in[4];   // [1, 512]
    const float* Wv_b = (const float*)d_in[5];   // [1]
    (void)in_sizes; (void)n_in; (void)out_size; (void)ws_size;

    // workspace layout
    float* Kmat = (float*)d_ws;                              // 8192*512
    float* V    = Kmat + (size_t)N_NODES * C_DIM;            // 8192
    float* yv   = V + N_NODES;                               // 8192
    int*   rank = (int*)(yv + N_NODES);                      // 8192
    int*   idx  = rank + N_NODES;                            // 2048

    // output layout: A_pooled | X_pooled | indices(as f32)
    float* out      = (float*)d_out;
    float* A_pooled = out;                                         // 2048*2048
    float* X_pooled = out + (size_t)K_SEL * K_SEL;                 // 2048*512
    float* out_idx  = X_pooled + (size_t)K_SEL * C_DIM;            // 2048

    // 1) K = X Wk^T + b   (f32 WMMA, 16x64 strip per wave)
    wk_gemm_kernel<<<dim3(N_NODES / 16, C_DIM / 64), 32, 0, stream>>>(X, Wk_w, Wk_b, Kmat);
    // 2) V = X Wv^T + b
    wv_gemv_kernel<<<N_NODES / 8, 256, 0, stream>>>(X, Wv_w, Wv_b, V);
    // 3) y = softmax(X K^T) V   (fused, f32 WMMA + online softmax)
    attn_y_kernel<<<N_NODES / 16, 256, 0, stream>>>(X, Kmat, V, yv);
    // 4) exact top-k ranks
    rank_kernel<<<N_NODES / 256, 256, 0, stream>>>(yv, rank);
    // 5) compact to sorted indices
    compact_kernel<<<N_NODES / 256, 256, 0, stream>>>(rank, idx, out_idx);
    // 6) gathers
    gather_x_kernel<<<K_SEL, 256, 0, stream>>>(X, idx, X_pooled);
    gather_a_kernel<<<dim3(K_SEL / 256, K_SEL), 256, 0, stream>>>(A, idx, A_pooled);
}